// _DenseGATConv_8830452761366
// MI455X (gfx1250) — compile-verified
//
#include <hip/hip_runtime.h>
#include <cstdint>

typedef float v2f __attribute__((ext_vector_type(2)));
typedef float v4f __attribute__((ext_vector_type(4)));
typedef float v8f __attribute__((ext_vector_type(8)));

#define WMMA_F32_4(A, B, C) \
  __builtin_amdgcn_wmma_f32_16x16x4_f32(false, (A), false, (B), (short)0, (C), false, false)

// CDNA5 async memory->LDS (VGLOBAL encoding, tracked by ASYNCcnt).
// lds_off: byte offset within workgroup LDS (low 32 bits of generic shared addr).
__device__ __forceinline__ void async_b128(uint32_t lds_off, const float* g) {
  asm volatile("global_load_async_to_lds_b128 %0, %1, off"
               :: "v"(lds_off), "v"((unsigned long long)(uintptr_t)g)
               : "memory");
}
#define LDS_OFF(p) ((uint32_t)(uintptr_t)(p))
#define S_WAIT_ASYNCCNT(n) asm volatile("s_wait_asynccnt " #n ::: "memory")

constexpr int Bn = 8;
constexpr int Nn = 1024;
constexpr int Cc = 256;   // input channels
constexpr int Hh = 4;
constexpr int Dd = 64;
constexpr int HD = 256;   // H*D
constexpr float NEG_SLOPE = 0.2f;

// ---------------------------------------------------------------------------
// K1: Wx = x @ W (per batch 1024x256 @ 256x256), fp32 WMMA 16x16x4.
// grid (N/64, HD/64, B), block 256. Async double-buffered LDS staging.
// ---------------------------------------------------------------------------
__global__ __launch_bounds__(256) void k_gemm_xw(const float* __restrict__ x,
                                                 const float* __restrict__ W,
                                                 float* __restrict__ Wx) {
  __shared__ __align__(16) float xs[2][64][20];  // 64 n x 16 k (stride 20: 16B-aligned, conflict-free)
  __shared__ __align__(16) float wt[2][16][72];  // 16 k x 64 c (stride 72)
  const int tid  = threadIdx.x;
  const int lane = tid & 31;
  const int wave = tid >> 5;
  const int half = lane >> 4;
  const int lm   = lane & 15;
  const int n0 = blockIdx.x * 64;
  const int c0 = blockIdx.y * 64;
  const int b  = blockIdx.z;
  const int wr = wave & 3;   // 4 row groups of 16
  const int wc = wave >> 2;  // 2 col groups of 32

  // per-thread async chunk coordinates (one b128 per tile per array)
  const int xr = tid >> 2, xc = (tid & 3) * 4;     // 64x16 tile: 256 chunks
  const int wrr = tid >> 4, wcc = (tid & 15) * 4;  // 16x64 tile: 256 chunks

  const float* xbase = x + (size_t)(b * Nn + n0) * Cc;

  v8f acc0, acc1;
#pragma unroll
  for (int r = 0; r < 8; ++r) { acc0[r] = 0.0f; acc1[r] = 0.0f; }

  // prefetch tile 0
  async_b128(LDS_OFF(&xs[0][xr][xc]), xbase + (size_t)xr * Cc + xc);
  async_b128(LDS_OFF(&wt[0][wrr][wcc]), W + (size_t)wrr * HD + c0 + wcc);

  int buf = 0;
  for (int k0 = 0; k0 < Cc; k0 += 16, buf ^= 1) {
    if (k0 + 16 < Cc) {
      async_b128(LDS_OFF(&xs[buf ^ 1][xr][xc]), xbase + (size_t)xr * Cc + (k0 + 16) + xc);
      async_b128(LDS_OFF(&wt[buf ^ 1][wrr][wcc]), W + (size_t)(k0 + 16 + wrr) * HD + c0 + wcc);
      S_WAIT_ASYNCCNT(2);   // current tile done (in-order), next still in flight
    } else {
      S_WAIT_ASYNCCNT(0);
    }
    __syncthreads();
#pragma unroll
    for (int ks = 0; ks < 16; ks += 4) {
      // A frag (16x4 f32): lanes 0-15 K={ks,ks+1}, lanes 16-31 K={ks+2,ks+3}
      v2f av, b0, b1;
      av.x = xs[buf][wr * 16 + lm][ks + 2 * half];
      av.y = xs[buf][wr * 16 + lm][ks + 2 * half + 1];
      b0.x = wt[buf][ks + 2 * half][wc * 32 + lm];
      b0.y = wt[buf][ks + 2 * half + 1][wc * 32 + lm];
      b1.x = wt[buf][ks + 2 * half][wc * 32 + 16 + lm];
      b1.y = wt[buf][ks + 2 * half + 1][wc * 32 + 16 + lm];
      acc0 = WMMA_F32_4(av, b0, acc0);
      acc1 = WMMA_F32_4(av, b1, acc1);
    }
    __syncthreads();   // everyone done reading buf before it is overwritten
  }

  // C layout: VGPR r -> row (r + 8*half), col = lane%16
  float* orow = Wx + (size_t)(b * Nn + n0 + wr * 16 + half * 8) * HD + c0 + wc * 32 + lm;
#pragma unroll
  for (int r = 0; r < 8; ++r) {
    orow[(size_t)r * HD]      = acc0[r];
    orow[(size_t)r * HD + 16] = acc1[r];
  }
}

// ---------------------------------------------------------------------------
// K2a: e_i[b,n,h] = Wx[b,n,h,:]·a_i[h], e_j likewise. One wave per (b,n).
// ---------------------------------------------------------------------------
__global__ __launch_bounds__(256) void k_edge(const float* __restrict__ Wx,
                                              const float* __restrict__ a,
                                              float* __restrict__ ei,
                                              float* __restrict__ ej) {
  const int tid  = threadIdx.x;
  const int lane = tid & 31;
  const int wave = tid >> 5;
  const int wg   = blockIdx.x * 8 + wave;  // == b*N + n
  const float* row = Wx + (size_t)wg * HD;
  const int h  = lane >> 3;
  const int d0 = (lane & 7) * 8;
  v4f v0 = *(const v4f*)(row + lane * 8);
  v4f v1 = *(const v4f*)(row + lane * 8 + 4);
  const float* ai = a + h * (2 * Dd) + d0;
  const float* aj = a + h * (2 * Dd) + Dd + d0;
  float si = 0.0f, sj = 0.0f;
#pragma unroll
  for (int t = 0; t < 4; ++t) { si += v0[t] * ai[t];     sj += v0[t] * aj[t]; }
#pragma unroll
  for (int t = 0; t < 4; ++t) { si += v1[t] * ai[4 + t]; sj += v1[t] * aj[4 + t]; }
#pragma unroll
  for (int off = 4; off >= 1; off >>= 1) {
    si += __shfl_xor(si, off, 8);
    sj += __shfl_xor(sj, off, 8);
  }
  if ((lane & 7) == 0) {
    ei[(size_t)wg * Hh + h] = si;
    ej[(size_t)wg * Hh + h] = sj;
  }
}

// ---------------------------------------------------------------------------
// K2b: per-(b,n,h) masked softmax stats (running max m, sum l) over j.
// ---------------------------------------------------------------------------
__global__ __launch_bounds__(256) void k_stats(const float* __restrict__ adj,
                                               const float* __restrict__ ei,
                                               const float* __restrict__ ej,
                                               float* __restrict__ mrow,
                                               float* __restrict__ lrow) {
  const int tid  = threadIdx.x;
  const int lane = tid & 31;
  const int wave = tid >> 5;
  const int wg   = blockIdx.x * 8 + wave;  // b*N + n
  const int n    = wg & (Nn - 1);
  const int bN   = wg - n;                 // b*N
  const float* arow = adj + (size_t)wg * Nn;
  v4f e_i = *(const v4f*)(ei + (size_t)wg * Hh);

  float m[4], s[4];
#pragma unroll
  for (int h = 0; h < 4; ++h) { m[h] = -INFINITY; s[h] = 0.0f; }

  for (int j = lane; j < Nn; j += 32) {
    float av = arow[j];
    if (av != 0.0f || j == n) {
      v4f e4 = *(const v4f*)(ej + (size_t)(bN + j) * Hh);
#pragma unroll
      for (int h = 0; h < 4; ++h) {
        float v = e_i[h] + e4[h];
        v = v > 0.0f ? v : NEG_SLOPE * v;
        if (v > m[h]) { s[h] = s[h] * __expf(m[h] - v) + 1.0f; m[h] = v; }
        else          { s[h] += __expf(v - m[h]); }
      }
    }
  }
#pragma unroll
  for (int off = 16; off >= 1; off >>= 1) {
#pragma unroll
    for (int h = 0; h < 4; ++h) {
      float m2 = __shfl_xor(m[h], off, 32);
      float s2 = __shfl_xor(s[h], off, 32);
      float nm = fmaxf(m[h], m2);
      if (nm > -INFINITY) {
        s[h] = s[h] * __expf(m[h] - nm) + s2 * __expf(m2 - nm);
        m[h] = nm;
      }
    }
  }
  if (lane == 0) {
#pragma unroll
    for (int h = 0; h < 4; ++h) {
      mrow[(size_t)wg * Hh + h] = m[h];
      lrow[(size_t)wg * Hh + h] = s[h];
    }
  }
}

// ---------------------------------------------------------------------------
// K3: out[b,n,h,:] = sum_j alpha(n,j) * Wx[b,j,h,:], alpha built on the fly.
// grid (N/128, H, B), block 256 (8 waves). Wave w owns rows n0+16w..+15,
// four 16x16 f32 WMMA accumulators per wave. Async double-buffered tiles
// of 32 j-columns (Wx 32x64, adj 128x32).
// ---------------------------------------------------------------------------
__global__ __launch_bounds__(256) void k_aggr(const float* __restrict__ adj,
                                              const float* __restrict__ Wx,
                                              const float* __restrict__ ei,
                                              const float* __restrict__ ej,
                                              const float* __restrict__ mrow,
                                              const float* __restrict__ lrow,
                                              float* __restrict__ out) {
  __shared__ __align__(16) float wx_s[2][32][72];   // j x d (stride 72: aligned, conflict-free)
  __shared__ __align__(16) float adj_s[2][128][36]; // n x j (stride 36)
  __shared__ float ej_s[2][32];

  const int tid  = threadIdx.x;
  const int lane = tid & 31;
  const int wave = tid >> 5;
  const int half = lane >> 4;
  const int lm   = lane & 15;
  const int n0 = blockIdx.x * 128;
  const int h  = blockIdx.y;
  const int b  = blockIdx.z;
  const int row_l = wave * 16 + lm;
  const int n_row = n0 + row_l;
  const size_t bN = (size_t)b * Nn;

  const float e_i_r = ei[(bN + n_row) * Hh + h];
  const float m_r   = mrow[(bN + n_row) * Hh + h];
  const float inv_l = 1.0f / lrow[(bN + n_row) * Hh + h];

  // per-thread async chunk coordinates
  const int wxr = tid >> 4, wxc = (tid & 15) * 4;  // 32x64 Wx tile: 512 chunks -> 2/thread
  const int adr = tid >> 3, adc = (tid & 7) * 4;   // 128x32 adj tile: 1024 chunks -> 4/thread

  const float* wx_g  = Wx + bN * HD + h * Dd;
  const float* adj_g = adj + (bN + n0) * Nn;

  v8f acc[4];
#pragma unroll
  for (int c = 0; c < 4; ++c)
#pragma unroll
    for (int r = 0; r < 8; ++r) acc[c][r] = 0.0f;

  // prefetch tile 0 (6 async issues per wave per tile: 2 Wx + 4 adj)
#pragma unroll
  for (int i = 0; i < 2; ++i)
    async_b128(LDS_OFF(&wx_s[0][wxr + i * 16][wxc]),
               wx_g + (size_t)(wxr + i * 16) * HD + wxc);
#pragma unroll
  for (int i = 0; i < 4; ++i)
    async_b128(LDS_OFF(&adj_s[0][adr + i * 32][adc]),
               adj_g + (size_t)(adr + i * 32) * Nn + adc);

  int buf = 0;
  for (int jt = 0; jt < Nn; jt += 32, buf ^= 1) {
    if (jt + 32 < Nn) {
#pragma unroll
      for (int i = 0; i < 2; ++i)
        async_b128(LDS_OFF(&wx_s[buf ^ 1][wxr + i * 16][wxc]),
                   wx_g + (size_t)(jt + 32 + wxr + i * 16) * HD + wxc);
#pragma unroll
      for (int i = 0; i < 4; ++i)
        async_b128(LDS_OFF(&adj_s[buf ^ 1][adr + i * 32][adc]),
                   adj_g + (size_t)(adr + i * 32) * Nn + jt + 32 + adc);
      S_WAIT_ASYNCCNT(6);   // tile jt complete (in-order), tile jt+32 in flight
    } else {
      S_WAIT_ASYNCCNT(0);
    }
    if (tid < 32) ej_s[buf][tid] = ej[(bN + jt + tid) * Hh + h];
    __syncthreads();

#pragma unroll
    for (int ks = 0; ks < 8; ++ks) {
      const int j0 = ks * 4 + 2 * half;  // lane's K pair within tile
      v2f av;
      {
        float a0 = adj_s[buf][row_l][j0];
        float a1 = adj_s[buf][row_l][j0 + 1];
        bool msk0 = (a0 != 0.0f) || (n_row == jt + j0);
        bool msk1 = (a1 != 0.0f) || (n_row == jt + j0 + 1);
        float v0 = e_i_r + ej_s[buf][j0];
        float v1 = e_i_r + ej_s[buf][j0 + 1];
        v0 = v0 > 0.0f ? v0 : NEG_SLOPE * v0;
        v1 = v1 > 0.0f ? v1 : NEG_SLOPE * v1;
        av.x = msk0 ? __expf(v0 - m_r) * inv_l : 0.0f;
        av.y = msk1 ? __expf(v1 - m_r) * inv_l : 0.0f;
      }
#pragma unroll
      for (int c = 0; c < 4; ++c) {
        v2f bv;
        bv.x = wx_s[buf][j0][c * 16 + lm];
        bv.y = wx_s[buf][j0 + 1][c * 16 + lm];
        acc[c] = WMMA_F32_4(av, bv, acc[c]);
      }
    }
    __syncthreads();   // all waves done with buf before overwrite
  }

  float* orow = out + (bN + n0 + wave * 16 + half * 8) * HD + h * Dd + lm;
#pragma unroll
  for (int c = 0; c < 4; ++c)
#pragma unroll
    for (int r = 0; r < 8; ++r)
      orow[(size_t)r * HD + c * 16] = acc[c][r];
}

// ---------------------------------------------------------------------------
extern "C" void kernel_launch(void* const* d_in, const int* in_sizes, int n_in,
                              void* d_out, int out_size, void* d_ws, size_t ws_size,
                              hipStream_t stream) {
  const float* x   = (const float*)d_in[0];  // (8,1024,256)
  const float* adj = (const float*)d_in[1];  // (8,1024,1024)
  const float* W   = (const float*)d_in[2];  // (256,256)
  const float* a   = (const float*)d_in[3];  // (1,4,128)
  float* out = (float*)d_out;                // (8,1024,256)

  float* Wx = (float*)d_ws;                        // B*N*HD  = 2,097,152 f
  float* ei = Wx + (size_t)Bn * Nn * HD;           // B*N*H   = 32,768 f each
  float* ej = ei + (size_t)Bn * Nn * Hh;
  float* mr = ej + (size_t)Bn * Nn * Hh;
  float* lr = mr + (size_t)Bn * Nn * Hh;

  k_gemm_xw<<<dim3(Nn / 64, HD / 64, Bn), 256, 0, stream>>>(x, W, Wx);
  k_edge  <<<dim3((Bn * Nn) / 8), 256, 0, stream>>>(Wx, a, ei, ej);
  k_stats <<<dim3((Bn * Nn) / 8), 256, 0, stream>>>(adj, ei, ej, mr, lr);
  k_aggr  <<<dim3(Nn / 128, Hh, Bn), 256, 0, stream>>>(adj, Wx, ei, ej, mr, lr, out);
}